// CondensedEmbracementLayer_79456894976128
// MI455X (gfx1250) — compile-verified
//
#include <hip/hip_runtime.h>
#include <hip/hip_bf16.h>
#include <stdint.h>

#define BS   32
#define SEQ  4096
#define HID  1024
#define HALF_N ((BS * HID) / 2)   // 16384: threefry counter pairing distance

// ---- JAX threefry2x32, key = (0, 42)  (jax.random.key(42)) -----------------
__device__ __forceinline__ void tf_round(uint32_t& v0, uint32_t& v1, int r) {
  v0 += v1;
  v1 = (v1 << r) | (v1 >> (32 - r));
  v1 ^= v0;
}

__device__ __forceinline__ void threefry2x32_k0_42(uint32_t x0, uint32_t x1,
                                                   uint32_t& o0, uint32_t& o1) {
  const uint32_t ks0 = 0u;
  const uint32_t ks1 = 42u;
  const uint32_t ks2 = ks0 ^ ks1 ^ 0x1BD11BDAu;
  uint32_t v0 = x0 + ks0;
  uint32_t v1 = x1 + ks1;
  tf_round(v0, v1, 13); tf_round(v0, v1, 15); tf_round(v0, v1, 26); tf_round(v0, v1, 6);
  v0 += ks1; v1 += ks2 + 1u;
  tf_round(v0, v1, 17); tf_round(v0, v1, 29); tf_round(v0, v1, 16); tf_round(v0, v1, 24);
  v0 += ks2; v1 += ks0 + 2u;
  tf_round(v0, v1, 13); tf_round(v0, v1, 15); tf_round(v0, v1, 26); tf_round(v0, v1, 6);
  v0 += ks0; v1 += ks1 + 3u;
  tf_round(v0, v1, 17); tf_round(v0, v1, 29); tf_round(v0, v1, 16); tf_round(v0, v1, 24);
  v0 += ks1; v1 += ks2 + 4u;
  tf_round(v0, v1, 13); tf_round(v0, v1, 15); tf_round(v0, v1, 26); tf_round(v0, v1, 6);
  v0 += ks2; v1 += ks0 + 5u;
  o0 = v0; o1 = v1;
}

// One block per batch row. Fused: mask-prefix reduction + threefry + gather.
__global__ __launch_bounds__(HID) void condensed_embrace_kernel(
    const float* __restrict__ tokens,   // [BS, SEQ, HID] f32
    const int*   __restrict__ mask,     // [BS, SEQ] i32 (1s then 0s)
    float*       __restrict__ out) {    // [BS, HID] f32
  __shared__ __align__(16) int s_mask[SEQ];   // 16 KB of the 320 KB WGP LDS
  __shared__ int s_first;

  const int b   = blockIdx.x;
  const int tid = threadIdx.x;
  const int* mrow = mask + (size_t)b * SEQ;

  if (tid == 0) s_first = SEQ;

  // gfx1250 path: prefetch + async global->LDS copy (ASYNCcnt domain).
  __builtin_prefetch(mrow + tid * 4, 0, 0);   // -> global_prefetch_b8
#if defined(__gfx1250__)
  {
    typedef int v4i __attribute__((ext_vector_type(4)));
    typedef __attribute__((address_space(1))) v4i gv4i_t;
    typedef __attribute__((address_space(3))) v4i lv4i_t;
    __builtin_amdgcn_global_load_async_to_lds_b128(
        (gv4i_t*)(mrow + tid * 4),
        (lv4i_t*)(&s_mask[tid * 4]),
        /*offset=*/0, /*cpol=*/0);
  }
  asm volatile("s_wait_asynccnt 0" ::: "memory");
#else
  // Host-pass / non-gfx1250 parse fallback (never executed on device).
  {
    int4 v = *(const int4*)(mrow + tid * 4);
    *(int4*)(&s_mask[tid * 4]) = v;
  }
#endif
  __syncthreads();

  // First-zero index of this batch's mask row (default SEQ) == jnp.where/argmin.
  {
    const int base = tid * 4;
    int4 m = *(const int4*)(&s_mask[base]);
    int local = SEQ;
    if (m.w == 0) local = base + 3;
    if (m.z == 0) local = base + 2;
    if (m.y == 0) local = base + 1;
    if (m.x == 0) local = base + 0;
    if (local < SEQ) atomicMin(&s_first, local);
  }
  __syncthreads();

  const int prefix  = s_first;
  int n_valid = prefix - 1;
  if (n_valid < 1) n_valid = 1;

  // JAX uniform bits for flat element j; counters split into halves.
  const uint32_t j = (uint32_t)(b * HID + tid);
  uint32_t o0, o1, bits;
  if (j < (uint32_t)HALF_N) {
    threefry2x32_k0_42(j, j + (uint32_t)HALF_N, o0, o1);
    bits = o0;
  } else {
    threefry2x32_k0_42(j - (uint32_t)HALF_N, j, o0, o1);
    bits = o1;
  }
  float u = __uint_as_float((bits >> 9) | 0x3F800000u) - 1.0f;  // [0,1)
  u = fmaxf(u, 0.0f);

  int idx = (int)(u * (float)n_valid);          // trunc-toward-zero, u >= 0
  const int nm1 = n_valid - 1;
  if (idx > nm1) idx = nm1;

  out[(size_t)b * HID + tid] =
      tokens[((size_t)b * SEQ + (size_t)idx) * (size_t)HID + (size_t)tid];
}

extern "C" void kernel_launch(void* const* d_in, const int* in_sizes, int n_in,
                              void* d_out, int out_size, void* d_ws, size_t ws_size,
                              hipStream_t stream) {
  (void)in_sizes; (void)n_in; (void)out_size; (void)d_ws; (void)ws_size;
  const float* tokens = (const float*)d_in[0];   // [32,4096,1024] f32
  const int*   mask   = (const int*)d_in[1];     // [32,4096] i32
  float*       out    = (float*)d_out;           // [32,1024] f32

  condensed_embrace_kernel<<<dim3(BS), dim3(HID), 0, stream>>>(tokens, mask, out);
}